// AdaptiveBlock_85598698209885
// MI455X (gfx1250) — compile-verified
//
#include <hip/hip_runtime.h>
#include <math.h>

typedef float v2f __attribute__((ext_vector_type(2)));
typedef float v4f __attribute__((ext_vector_type(4)));
typedef float v8f __attribute__((ext_vector_type(8)));

#define BDIM 32
#define CDIM 1024
#define HIDDIM 512
#define HDIM 56
#define WDIM 56
#define HW (HDIM * WDIM)          // 3136
#define HW4 (HW / 4)              // 784 float4 per map

__device__ __forceinline__ float gelu_exact(float v) {
    return 0.5f * v * (1.0f + erff(v * 0.70710678118654752440f));
}
__device__ __forceinline__ float sigmoidf(float v) {
    return 1.0f / (1.0f + expf(-v));
}

// ---------------------------------------------------------------------------
// Kernel 1: global average pool.  One 256-thread block per (b,c) channel.
// Streaming read of 411 MB -> nontemporal b128 loads.
// ---------------------------------------------------------------------------
__global__ void pool_kernel(const float* __restrict__ x, float* __restrict__ y) {
    const int bc = blockIdx.x;                       // [0, B*C)
    const v4f* p4 = (const v4f*)(x + (size_t)bc * HW);
    float sum = 0.0f;
    for (int i = threadIdx.x; i < HW4; i += 256) {
        v4f v = __builtin_nontemporal_load(&p4[i]);
        sum += (v.x + v.y) + (v.z + v.w);
    }
    // wave32 reduction
    for (int off = 16; off > 0; off >>= 1)
        sum += __shfl_xor(sum, off, 32);
    __shared__ float wsum[8];
    const int wave = threadIdx.x >> 5;
    if ((threadIdx.x & 31) == 0) wsum[wave] = sum;
    __syncthreads();
    if (threadIdx.x == 0) {
        float t = 0.0f;
        #pragma unroll
        for (int i = 0; i < 8; ++i) t += wsum[i];
        y[bc] = t * (1.0f / (float)HW);
    }
}

// ---------------------------------------------------------------------------
// WMMA helper: one wave computes a 16x16 f32 tile of A(MxK) * B(KxN),
// both row-major, using V_WMMA_F32_16X16X4_F32 (K-step 4).
// A frag: lanes 0-15 K={k0,k0+1}; lanes 16-31 K={k0+2,k0+3}; M = lane%16.
// B frag: same K mapping; N = n0 + lane%16.
// ---------------------------------------------------------------------------
__device__ __forceinline__ v8f wmma_tile_f32(const float* __restrict__ A, int lda,
                                             const float* __restrict__ B, int ldb,
                                             int K, int m0, int n0, int lane) {
    const int half = lane >> 4;
    const int l    = lane & 15;
    v8f c = {};
    for (int k0 = 0; k0 < K; k0 += 4) {
        const int ka = k0 + 2 * half;
        v2f av, bv;
        av.x = A[(m0 + l) * lda + ka];
        av.y = A[(m0 + l) * lda + ka + 1];
        bv.x = B[ka * ldb + (n0 + l)];
        bv.y = B[(ka + 1) * ldb + (n0 + l)];
        c = __builtin_amdgcn_wmma_f32_16x16x4_f32(false, av, false, bv,
                                                  (short)0, c, false, false);
    }
    return c;
}

// ---------------------------------------------------------------------------
// Kernel 2: h = gelu(y @ W1)   (32x1024)(1024x512) -> 32x512
// 2 M-tiles x 32 N-tiles = 64 waves -> 8 blocks x 256 threads
// ---------------------------------------------------------------------------
__global__ void gemm1_kernel(const float* __restrict__ y, const float* __restrict__ W1,
                             float* __restrict__ h) {
    const int gwave = (blockIdx.x * blockDim.x + threadIdx.x) >> 5;
    const int lane  = threadIdx.x & 31;
    const int mt = gwave >> 5;            // 0..1
    const int nt = gwave & 31;            // 0..31
    const int m0 = mt * 16, n0 = nt * 16;
    v8f c = wmma_tile_f32(y, CDIM, W1, HIDDIM, CDIM, m0, n0, lane);
    const int half = lane >> 4, l = lane & 15;
    #pragma unroll
    for (int r = 0; r < 8; ++r) {
        const int m = m0 + r + 8 * half;
        h[m * HIDDIM + (n0 + l)] = gelu_exact(c[r]);
    }
}

// ---------------------------------------------------------------------------
// Kernel 3: yp = gelu(h @ W2)  (32x512)(512x1024) -> 32x1024
// 2 x 64 = 128 waves -> 16 blocks x 256 threads
// ---------------------------------------------------------------------------
__global__ void gemm2_kernel(const float* __restrict__ h, const float* __restrict__ W2,
                             float* __restrict__ yp) {
    const int gwave = (blockIdx.x * blockDim.x + threadIdx.x) >> 5;
    const int lane  = threadIdx.x & 31;
    const int mt = gwave >> 6;            // 0..1
    const int nt = gwave & 63;            // 0..63
    const int m0 = mt * 16, n0 = nt * 16;
    v8f c = wmma_tile_f32(h, HIDDIM, W2, CDIM, HIDDIM, m0, n0, lane);
    const int half = lane >> 4, l = lane & 15;
    #pragma unroll
    for (int r = 0; r < 8; ++r) {
        const int m = m0 + r + 8 * half;
        yp[m * CDIM + (n0 + l)] = gelu_exact(c[r]);
    }
}

// ---------------------------------------------------------------------------
// Kernel 4: Av = yp @ WA + bA ; Bv = yp @ WB + bB   (32x1024)(1024x56)
// N=56 padded to 64 in-tile (predicated loads/stores, EXEC stays full for WMMA).
// block 0 -> Av, block 1 -> Bv; 8 waves = 2 M-tiles x 4 N-tiles.
// ---------------------------------------------------------------------------
__global__ void gemm3_kernel(const float* __restrict__ yp,
                             const float* __restrict__ WA, const float* __restrict__ bA,
                             const float* __restrict__ WB, const float* __restrict__ bB,
                             float* __restrict__ Av, float* __restrict__ Bv) {
    const float* Wm   = (blockIdx.x == 0) ? WA : WB;
    const float* bias = (blockIdx.x == 0) ? bA : bB;
    float*       outp = (blockIdx.x == 0) ? Av : Bv;

    const int wave = threadIdx.x >> 5;    // 0..7
    const int lane = threadIdx.x & 31;
    const int mt = wave >> 2;             // 0..1
    const int nt = wave & 3;              // 0..3
    const int m0 = mt * 16, n0 = nt * 16;
    const int half = lane >> 4, l = lane & 15;
    const int n = n0 + l;
    const bool valid = (n < HDIM);        // HDIM == WDIM == 56

    v8f c = {};
    for (int k0 = 0; k0 < CDIM; k0 += 4) {
        const int ka = k0 + 2 * half;
        v2f av, bv;
        av.x = yp[(m0 + l) * CDIM + ka];
        av.y = yp[(m0 + l) * CDIM + ka + 1];
        bv.x = valid ? Wm[ka * HDIM + n] : 0.0f;
        bv.y = valid ? Wm[(ka + 1) * HDIM + n] : 0.0f;
        c = __builtin_amdgcn_wmma_f32_16x16x4_f32(false, av, false, bv,
                                                  (short)0, c, false, false);
    }
    if (valid) {
        const float bn = bias[n];
        #pragma unroll
        for (int r = 0; r < 8; ++r) {
            const int m = m0 + r + 8 * half;
            outp[m * HDIM + n] = c[r] + bn;
        }
    }
}

// ---------------------------------------------------------------------------
// Kernel 5: map[b,i,j] = sigmoid(Av[b,i] * Bv[b,j])   (32 x 56 x 56)
// ---------------------------------------------------------------------------
__global__ void attnmap_kernel(const float* __restrict__ Av, const float* __restrict__ Bv,
                               float* __restrict__ map) {
    const int idx = blockIdx.x * 256 + threadIdx.x;
    if (idx < BDIM * HW) {
        const int b = idx / HW;
        const int r = idx % HW;
        const int i = r / WDIM;
        const int j = r % WDIM;
        map[idx] = sigmoidf(Av[b * HDIM + i] * Bv[b * WDIM + j]);
    }
}

// ---------------------------------------------------------------------------
// Kernel 6: broadcast map across channels.  Block = (b, group of 8 channels).
// Map tile held in registers (4 x float4 / thread); NT stores (write-once data).
// ---------------------------------------------------------------------------
__global__ void bcast_kernel(const float* __restrict__ map, float* __restrict__ out) {
    const int b  = blockIdx.x >> 7;                   // [0,32)
    const int cg = blockIdx.x & 127;                  // [0,128) * 8 channels
    const int t  = threadIdx.x;
    const v4f* m4 = (const v4f*)(map + (size_t)b * HW);

    v4f r0 = m4[t];
    v4f r1 = m4[t + 256];
    v4f r2 = m4[t + 512];
    v4f r3 = {};
    const bool has4 = (t < HW4 - 768);                // 784 - 768 = 16 tail vecs
    if (has4) r3 = m4[t + 768];

    const size_t base = ((size_t)b * CDIM + (size_t)cg * 8) * HW;
    #pragma unroll
    for (int c = 0; c < 8; ++c) {
        v4f* o4 = (v4f*)(out + base + (size_t)c * HW);
        __builtin_nontemporal_store(r0, &o4[t]);
        __builtin_nontemporal_store(r1, &o4[t + 256]);
        __builtin_nontemporal_store(r2, &o4[t + 512]);
        if (has4) __builtin_nontemporal_store(r3, &o4[t + 768]);
    }
}

// ---------------------------------------------------------------------------
extern "C" void kernel_launch(void* const* d_in, const int* in_sizes, int n_in,
                              void* d_out, int out_size, void* d_ws, size_t ws_size,
                              hipStream_t stream) {
    const float* x  = (const float*)d_in[0];   // (32,1024,56,56)
    const float* W1 = (const float*)d_in[1];   // (1024,512)
    const float* W2 = (const float*)d_in[2];   // (512,1024)
    const float* WA = (const float*)d_in[3];   // (1024,56)
    const float* bA = (const float*)d_in[4];   // (56)
    const float* WB = (const float*)d_in[5];   // (1024,56)
    const float* bB = (const float*)d_in[6];   // (56)
    float* out = (float*)d_out;

    float* y   = (float*)d_ws;                 // 32*1024
    float* h   = y   + BDIM * CDIM;            // 32*512
    float* yp  = h   + BDIM * HIDDIM;          // 32*1024
    float* Av  = yp  + BDIM * CDIM;            // 32*56
    float* Bv  = Av  + BDIM * HDIM;            // 32*56
    float* map = Bv  + BDIM * WDIM;            // 32*3136

    // 1) global average pool
    pool_kernel<<<BDIM * CDIM, 256, 0, stream>>>(x, y);
    // 2) h = gelu(y @ W1)       : 64 waves
    gemm1_kernel<<<8, 256, 0, stream>>>(y, W1, h);
    // 3) yp = gelu(h @ W2)      : 128 waves
    gemm2_kernel<<<16, 256, 0, stream>>>(h, W2, yp);
    // 4) Av / Bv                : 2 blocks x 8 waves
    gemm3_kernel<<<2, 256, 0, stream>>>(yp, WA, bA, WB, bB, Av, Bv);
    // 5) sigmoid outer product map
    attnmap_kernel<<<(BDIM * HW + 255) / 256, 256, 0, stream>>>(Av, Bv, map);
    // 6) broadcast across channels (write 411 MB, NT stores)
    bcast_kernel<<<BDIM * (CDIM / 8), 256, 0, stream>>>(map, out);
}